// KMeans_12171937317453
// MI455X (gfx1250) — compile-verified
//
#include <hip/hip_runtime.h>
#include <hip/hip_bf16.h>
#include <cfloat>

typedef __attribute__((ext_vector_type(2))) float v2f;
typedef __attribute__((ext_vector_type(8))) float v8f;

#define N_PTS 131072
#define K_CL  512
#define D_DIM 256

// ---------------------------------------------------------------------------
// ws layout (floats):
//   [0, 512)                : c2[k] = ||centroid_k||^2
//   [512, 512+K*D)          : sums[K,D]   (atomic accumulators, zeroed each call)
//   [512+K*D, 512+K*D+K)    : counts[K]
//   [512+K*D+K, +N)         : assignments as int32
// ---------------------------------------------------------------------------

__global__ void zero_ws_kernel(float* __restrict__ ws) {
  int i = blockIdx.x * blockDim.x + threadIdx.x;
  if (i < K_CL * D_DIM + K_CL) ws[512 + i] = 0.0f;
}

__global__ void c2_kernel(const float* __restrict__ c, float* __restrict__ c2) {
  int k = blockIdx.x * blockDim.x + threadIdx.x;
  if (k >= K_CL) return;
  const float4* cr = (const float4*)(c + (size_t)k * D_DIM);
  float s = 0.0f;
  #pragma unroll 8
  for (int j = 0; j < D_DIM / 4; ++j) {
    float4 v = cr[j];
    s += v.x * v.x + v.y * v.y + v.z * v.z + v.w * v.w;
  }
  c2[k] = s;
}

// ---------------------------------------------------------------------------
// Fused distance + argmin kernel.
// Block = 256 threads = 8 waves; each wave owns 16 rows of X.
// A-fragments (16x256 fp32 per wave = 64 x v2f per lane) are held entirely in
// VGPRs for the whole kernel; only the shared 16-centroid chunk (16 KB LDS)
// is re-staged per column chunk. GEMM via V_WMMA_F32_16X16X4_F32.
// score = c2[k] - 2*(x . c)  (same argmin as Euclidean distance).
// ---------------------------------------------------------------------------
__launch_bounds__(256)
__global__ void assign_kernel(const float* __restrict__ x,
                              const float* __restrict__ cent,
                              const float* __restrict__ c2,
                              int*   __restrict__ assign_i,
                              float* __restrict__ assign_f) {
  __shared__ float ctile[16 * D_DIM];   // 16 KB: current 16-centroid chunk

  const int t    = threadIdx.x;
  const int wave = t >> 5;
  const int lane = t & 31;
  const int l16  = lane & 15;
  const int hi   = lane >> 4;           // 0: lanes 0-15, 1: lanes 16-31

  const int rowbase = blockIdx.x * 128 + wave * 16;

  // fp32 A fragment layout (16x4 per WMMA): lane holds row l16,
  // K = 2*hi + {0,1}.  Load all 64 K-steps of A into registers once.
  v2f areg[64];
  {
    const float* xrow = x + ((size_t)rowbase + l16) * D_DIM + 2 * hi;
    #pragma unroll
    for (int j = 0; j < 64; ++j)
      areg[j] = *(const v2f*)(xrow + j * 4);
  }

  float minv[8];
  int   mini[8];
  #pragma unroll
  for (int v = 0; v < 8; ++v) { minv[v] = FLT_MAX; mini[v] = 0; }

  const float* brow = ctile + l16 * D_DIM + 2 * hi;  // B = C^T chunk fragment

  for (int cc = 0; cc < K_CL / 16; ++cc) {
    const int kbase = cc * 16;
    __syncthreads();
    // Stage centroid chunk [16 x 256] (4 float4 per thread, coalesced).
    {
      const float4* src = (const float4*)(cent + (size_t)kbase * D_DIM);
      float4* dst = (float4*)ctile;
      #pragma unroll
      for (int j = 0; j < 4; ++j) dst[t + j * 256] = src[t + j * 256];
    }
    __syncthreads();

    v8f acc;
    #pragma unroll
    for (int v = 0; v < 8; ++v) acc[v] = 0.0f;

    #pragma unroll
    for (int j = 0; j < 64; ++j) {
      v2f b = *(const v2f*)(brow + j * 4);
      acc = __builtin_amdgcn_wmma_f32_16x16x4_f32(
          /*neg_a=*/false, areg[j], /*neg_b=*/false, b,
          /*c_mod=*/(short)0, acc, /*reuse_a=*/false, /*reuse_b=*/false);
    }

    // score = ||c||^2 - 2 * dot ; this lane's column is kbase + l16.
    const float cc2    = c2[kbase + l16];
    const int   colIdx = kbase + l16;
    #pragma unroll
    for (int v = 0; v < 8; ++v) {
      float s = fmaf(-2.0f, acc[v], cc2);
      if (s < minv[v]) { minv[v] = s; mini[v] = colIdx; }   // strict <: keeps first
    }
  }

  // Cross-lane argmin over each 16-lane half (xor 8,4,2,1 stays within half).
  #pragma unroll
  for (int v = 0; v < 8; ++v) {
    float mv = minv[v]; int mi = mini[v];
    #pragma unroll
    for (int off = 8; off >= 1; off >>= 1) {
      float ov = __shfl_xor(mv, off, 32);
      int   oi = __shfl_xor(mi, off, 32);
      if (ov < mv || (ov == mv && oi < mi)) { mv = ov; mi = oi; }
    }
    minv[v] = mv; mini[v] = mi;
  }

  // Lane 0 holds rows M=0..7, lane 16 holds rows M=8..15 of this wave's tile.
  if (l16 == 0) {
    const int mbase = rowbase + hi * 8;
    #pragma unroll
    for (int v = 0; v < 8; ++v) {
      assign_i[mbase + v] = mini[v];
      assign_f[mbase + v] = (float)mini[v];
    }
  }
}

// ---------------------------------------------------------------------------
// Segment-sum via L2 float atomics. One float4 of one row per thread.
// ---------------------------------------------------------------------------
__global__ void scatter_kernel(const float* __restrict__ x,
                               const int* __restrict__ assign_i,
                               float* __restrict__ sums,
                               float* __restrict__ counts) {
  int gid = blockIdx.x * blockDim.x + threadIdx.x;   // N * (D/4) threads
  int r   = gid >> 6;                                 // D/4 == 64
  int d4  = gid & 63;
  int a   = assign_i[r];
  float4 v = ((const float4*)x)[gid];
  float* dst = sums + (size_t)a * D_DIM + d4 * 4;
  atomicAdd(dst + 0, v.x);
  atomicAdd(dst + 1, v.y);
  atomicAdd(dst + 2, v.z);
  atomicAdd(dst + 3, v.w);
  if (d4 == 0) atomicAdd(counts + a, 1.0f);
}

__global__ void finalize_kernel(const float* __restrict__ cent,
                                const float* __restrict__ sums,
                                const float* __restrict__ counts,
                                float* __restrict__ out_old,
                                float* __restrict__ out_new) {
  int i = blockIdx.x * blockDim.x + threadIdx.x;     // K * D
  int k = i >> 8;
  float cnt  = counts[k];
  float mean = sums[i] / fmaxf(cnt, 1.0f);
  float cold = cent[i];
  out_new[i] = (cnt > 0.0f) ? mean : cold;
  out_old[i] = cold;
}

// ---------------------------------------------------------------------------
extern "C" void kernel_launch(void* const* d_in, const int* in_sizes, int n_in,
                              void* d_out, int out_size, void* d_ws, size_t ws_size,
                              hipStream_t stream) {
  (void)in_sizes; (void)n_in; (void)out_size; (void)ws_size;

  const float* x    = (const float*)d_in[0];   // [N, D]
  const float* cent = (const float*)d_in[1];   // [K, D]
  float* out = (float*)d_out;
  float* ws  = (float*)d_ws;

  float* c2      = ws;
  float* sums    = ws + 512;
  float* counts  = ws + 512 + K_CL * D_DIM;
  int*   assignI = (int*)(ws + 512 + K_CL * D_DIM + K_CL);

  float* out_old    = out;                       // [1,K,D] old centroids
  float* out_assign = out + K_CL * D_DIM;        // [N] assignments (as float)
  float* out_new    = out + 2 * K_CL * D_DIM;    // [K,D] updated centroids

  zero_ws_kernel<<<(K_CL * D_DIM + K_CL + 255) / 256, 256, 0, stream>>>(ws);
  c2_kernel<<<(K_CL + 255) / 256, 256, 0, stream>>>(cent, c2);

  assign_kernel<<<N_PTS / 128, 256, 0, stream>>>(x, cent, c2, assignI, out_assign);

  scatter_kernel<<<(N_PTS * (D_DIM / 4)) / 256, 256, 0, stream>>>(x, assignI, sums, counts);
  finalize_kernel<<<(K_CL * D_DIM) / 256, 256, 0, stream>>>(cent, sums, counts,
                                                            out_old, out_new);
}